// InvariantPointAttention_2989297238038
// MI455X (gfx1250) — compile-verified
//
#include <hip/hip_runtime.h>
#include <math.h>

typedef __attribute__((ext_vector_type(2))) float v2f;
typedef __attribute__((ext_vector_type(8))) float v8f;

// ---------------------------------------------------------------------------
// Generic fp32 WMMA GEMM:  C[b] = alpha * A[b] @ B[b]  (+ per-column bias)
// V_WMMA_F32_16X16X4_F32. Block = 128 threads = 4 waves = 2x2 tiles (32x32).
// Batched via gridDim.z with per-operand strides.
// TRANS_B: B given as N x K row-major (logical B[k][n] = Bt[n*ldb + k]).
//
// Edge handling: OOB A rows / B cols are index-CLAMPED (not masked). The
// resulting garbage lands only in D rows/cols that the store skips, so the
// inner loop is branch-free: 2 x b64 loads + 1 wmma + pointer bumps, with a
// 2-stage software pipeline so next-step loads issue before the current WMMA.
// ---------------------------------------------------------------------------
template<bool TRANS_B>
__global__ __launch_bounds__(128)
void wmma_gemm(const float* __restrict__ A, long lda, long strideA,
               const float* __restrict__ B, long ldb, long strideB,
               const float* __restrict__ bias,
               float* __restrict__ C, long ldc, long strideC,
               int M, int N, int K, float alpha)
{
    const int wave = threadIdx.x >> 5;
    const int lane = threadIdx.x & 31;
    const int wm = wave >> 1, wn = wave & 1;
    const int m0 = blockIdx.x * 32 + wm * 16;
    const int n0 = blockIdx.y * 32 + wn * 16;
    if (m0 >= M || n0 >= N) return;           // wave-uniform exit (EXEC stays all-1)
    const long b = blockIdx.z;
    A += b * strideA;  B += b * strideB;  C += b * strideC;

    const int row = lane & 15;
    const int kh  = (lane >> 4) << 1;         // 0 or 2
    const int ar  = min(m0 + row, M - 1);     // clamped (see header comment)
    const int bc  = min(n0 + row, N - 1);

    const float* ap = A + (long)ar * lda + kh;
    const float* bp0;
    const float* bp1;
    long bstep;
    if (TRANS_B) {
        bp0 = B + (long)bc * ldb + kh;
        bp1 = bp0 + 1;
        bstep = 4;
    } else {
        bp0 = B + (long)kh * ldb + bc;
        bp1 = bp0 + ldb;
        bstep = 4 * ldb;
    }

    v8f acc = {};
    v2f a, bf;
    a.x  = ap[0];  a.y = ap[1];
    bf.x = bp0[0]; bf.y = bp1[0];
    for (int kk = 4; kk < K; kk += 4) {
        ap  += 4;
        bp0 += bstep;
        bp1 += bstep;
        v2f an, bn;
        an.x = ap[0];  an.y = ap[1];
        bn.x = bp0[0]; bn.y = bp1[0];
        acc = __builtin_amdgcn_wmma_f32_16x16x4_f32(false, a, false, bf,
                                                    (short)0, acc, false, false);
        a = an; bf = bn;
    }
    acc = __builtin_amdgcn_wmma_f32_16x16x4_f32(false, a, false, bf,
                                                (short)0, acc, false, false);

    const int rofs = (lane >> 4) << 3;        // +8 rows for upper half-wave
    const int col  = n0 + (lane & 15);
    if (col < N) {
        const float bv = bias ? bias[col] : 0.f;
        #pragma unroll
        for (int v = 0; v < 8; ++v) {
            const int r = m0 + v + rofs;
            if (r < M) C[(long)r * ldc + col] = alpha * acc[v] + bv;
        }
    }
}

// ---------------------------------------------------------------------------
// Apply frame: out[n,hp,:] = R[n] @ p[n,:,hp] + t[n]
// raw layout is component-major: [n][3][HP] (matches _pts reshape in ref).
// ---------------------------------------------------------------------------
__global__ void rotate_points(const float* __restrict__ raw,
                              const float* __restrict__ rot,
                              const float* __restrict__ trans,
                              float* __restrict__ out, int HP)
{
    int idx = blockIdx.x * blockDim.x + threadIdx.x;
    if (idx >= 768 * HP) return;
    int n = idx / HP, hp = idx % HP;
    const float* rb = raw + (long)n * 3 * HP;
    float px = rb[hp], py = rb[HP + hp], pz = rb[2 * HP + hp];
    const float* R = rot + (long)n * 9;
    const float* t = trans + (long)n * 3;
    float* o = out + (long)idx * 3;
    o[0] = R[0]*px + R[1]*py + R[2]*pz + t[0];
    o[1] = R[3]*px + R[4]*py + R[5]*pz + t[1];
    o[2] = R[6]*px + R[7]*py + R[8]*pz + t[2];
}

// ---------------------------------------------------------------------------
// Fused attention epilogue: per (row i, head h) block.
//   l = qk_scaled + sqrt(1/3)*b_bias + pt_term + mask_term
//   a = softmax_w(l), w = exp(ss) - 0.99   (in-place on att buffer)
// The raw qk row is staged global->LDS with the gfx1250 async-LDS engine
// (ASYNCcnt) instead of bouncing through VGPRs.
// ---------------------------------------------------------------------------
__global__ __launch_bounds__(256)
void attn_softmax(float* __restrict__ att, const float* __restrict__ bbias,
                  const float* __restrict__ qpts, const float* __restrict__ kvppts,
                  const float* __restrict__ head_weights,
                  const float* __restrict__ mask, const float* __restrict__ ss)
{
    const int i = blockIdx.x, h = blockIdx.y, tid = threadIdx.x;
    __shared__ float sl[768];
    __shared__ float red[256];

    // Async-stage the raw qk logits row into LDS (global_load_async_to_lds_b32,
    // tracked by ASYNCcnt), then fence with s_wait_asynccnt + barrier.
    float* att_row = att + (long)h * 589824 + (long)i * 768;
    for (int j = tid; j < 768; j += 256) {
        unsigned ldsa = (unsigned)(size_t)(&sl[j]);     // LDS byte offset
        const float* gp = att_row + j;
        asm volatile("global_load_async_to_lds_b32 %0, %1, off"
                     :: "v"(ldsa), "v"(gp) : "memory");
    }
    asm volatile("s_wait_asynccnt 0x0" ::: "memory");
    __syncthreads();

    // softplus(hw) * sqrt(1/(3*PQ*9/2)) = softplus * sqrt(1/54)
    const float hw = logf(1.f + expf(head_weights[h])) * 0.13608276348795434f;
    float qp[12];
    const float* qb = qpts + ((long)i * 48 + h * 4) * 3;
    #pragma unroll
    for (int t = 0; t < 12; ++t) qp[t] = qb[t];
    const float mi = mask[i];

    float lmax = -3.4e38f;
    for (int j = tid; j < 768; j += 256) {
        float l = sl[j]
                + 0.5773502691896258f * bbias[((long)i * 768 + j) * 12 + h];
        const float* kb = kvppts + ((long)j * 144 + h * 12) * 3;   // k_pts = p 0..3
        float sacc = 0.f;
        #pragma unroll
        for (int p = 0; p < 4; ++p) {
            float dx = qp[p*3+0] - kb[p*3+0];
            float dy = qp[p*3+1] - kb[p*3+1];
            float dz = qp[p*3+2] - kb[p*3+2];
            sacc += dx*dx + dy*dy + dz*dz;
        }
        l -= 0.5f * hw * sacc;
        l += 100000.0f * (mi * mask[j] - 1.f);
        sl[j] = l;
        lmax = fmaxf(lmax, l);
    }
    red[tid] = lmax; __syncthreads();
    for (int s2 = 128; s2 > 0; s2 >>= 1) {
        if (tid < s2) red[tid] = fmaxf(red[tid], red[tid + s2]);
        __syncthreads();
    }
    const float mx = red[0];
    __syncthreads();

    float lsum = 0.f;
    for (int j = tid; j < 768; j += 256) {
        float w = expf(ss[(long)i * 768 + j]) - 0.99f;
        float e = expf(sl[j] - mx) * w;
        sl[j] = e;
        lsum += e;
    }
    red[tid] = lsum; __syncthreads();
    for (int s2 = 128; s2 > 0; s2 >>= 1) {
        if (tid < s2) red[tid] += red[tid + s2];
        __syncthreads();
    }
    const float inv = 1.f / red[0];
    __syncthreads();
    for (int j = tid; j < 768; j += 256)
        att_row[j] = sl[j] * inv;
}

// o_pt back-rotation (R^T(x - t)), norm, write into feats cols [192,576)
__global__ void finalize_opt(const float* __restrict__ optb, const float* __restrict__ rot,
                             const float* __restrict__ trans, float* __restrict__ feats)
{
    int idx = blockIdx.x * blockDim.x + threadIdx.x;
    if (idx >= 768 * 96) return;
    int n = idx / 96, hp = idx % 96;
    int h = hp >> 3, p = hp & 7;
    const float* ob = optb + (long)n * 288 + h * 24 + p * 3;
    const float* t  = trans + (long)n * 3;
    float vx = ob[0] - t[0], vy = ob[1] - t[1], vz = ob[2] - t[2];
    const float* R = rot + (long)n * 9;
    float ox = R[0]*vx + R[3]*vy + R[6]*vz;   // R^T
    float oy = R[1]*vx + R[4]*vy + R[7]*vz;
    float oz = R[2]*vx + R[5]*vy + R[8]*vz;
    float nrm = sqrtf(ox*ox + oy*oy + oz*oz + 1e-8f);
    float* f = feats + (long)n * 2112;
    f[192 +   0 + hp] = ox;
    f[192 +  96 + hp] = oy;
    f[192 + 192 + hp] = oz;
    f[480 + hp] = nrm;
}

__global__ void copy_o(const float* __restrict__ o, float* __restrict__ feats)
{
    int idx = blockIdx.x * blockDim.x + threadIdx.x;
    if (idx >= 768 * 192) return;
    feats[(long)(idx / 192) * 2112 + (idx % 192)] = o[idx];
}

// ---------------------------------------------------------------------------
extern "C" void kernel_launch(void* const* d_in, const int* in_sizes, int n_in,
                              void* d_out, int out_size, void* d_ws, size_t ws_size,
                              hipStream_t stream)
{
    const float* s     = (const float*)d_in[0];
    const float* z     = (const float*)d_in[1];
    const float* rot   = (const float*)d_in[2];
    const float* trans = (const float*)d_in[3];
    const float* mask  = (const float*)d_in[4];
    const float* ss    = (const float*)d_in[5];
    const float* w_q   = (const float*)d_in[6];
    const float* b_q   = (const float*)d_in[7];
    const float* w_kv  = (const float*)d_in[8];
    const float* b_kv  = (const float*)d_in[9];
    const float* w_qp  = (const float*)d_in[10];
    const float* b_qp  = (const float*)d_in[11];
    const float* w_kvp = (const float*)d_in[12];
    const float* b_kvp = (const float*)d_in[13];
    const float* w_b   = (const float*)d_in[14];
    const float* b_b   = (const float*)d_in[15];
    const float* hwts  = (const float*)d_in[16];
    const float* w_out = (const float*)d_in[17];
    const float* b_out = (const float*)d_in[18];
    float* out = (float*)d_out;

    float* ws      = (float*)d_ws;
    float* q_buf   = ws;                  // 768*192
    float* kv_buf  = q_buf   + 147456;    // 768*384
    float* qpraw   = kv_buf  + 294912;    // 768*144
    float* kvpraw  = qpraw   + 110592;    // 768*432
    float* qpts    = kvpraw  + 331776;    // 768*48*3
    float* kvppts  = qpts    + 110592;    // 768*144*3
    float* bbias   = kvppts  + 331776;    // 768*768*12
    float* att     = bbias   + 7077888;   // 12*768*768
    float* o_buf   = att     + 7077888;   // 768*192
    float* opt_buf = o_buf   + 147456;    // 768*288
    float* feats   = opt_buf + 221184;    // 768*2112

    dim3 blk(128);
    auto tiles = [](int m, int n, int b) { return dim3((m + 31) / 32, (n + 31) / 32, b); };

    // 1. linear projections from s
    wmma_gemm<false><<<tiles(768,192,1), blk, 0, stream>>>(s,384,0, w_q,192,0,  b_q,  q_buf,192,0, 768,192,384, 1.f);
    wmma_gemm<false><<<tiles(768,384,1), blk, 0, stream>>>(s,384,0, w_kv,384,0, b_kv, kv_buf,384,0, 768,384,384, 1.f);
    wmma_gemm<false><<<tiles(768,144,1), blk, 0, stream>>>(s,384,0, w_qp,144,0, b_qp, qpraw,144,0, 768,144,384, 1.f);
    wmma_gemm<false><<<tiles(768,432,1), blk, 0, stream>>>(s,384,0, w_kvp,432,0,b_kvp,kvpraw,432,0, 768,432,384, 1.f);

    // 2. apply frames to points
    rotate_points<<<(768*48 +255)/256, 256, 0, stream>>>(qpraw,  rot, trans, qpts,   48);
    rotate_points<<<(768*144+255)/256, 256, 0, stream>>>(kvpraw, rot, trans, kvppts, 144);

    // 3. pair bias: (N*N x 128) @ (128 x 12)  -- first streaming pass over z
    wmma_gemm<false><<<tiles(589824,12,1), blk, 0, stream>>>(z,128,0, w_b,12,0, b_b, bbias,12,0, 589824,12,128, 1.f);

    // 4. scaled q.k^T logits, batched per head (K=16), alpha = sqrt(1/48)
    wmma_gemm<true><<<tiles(768,768,12), blk, 0, stream>>>(q_buf,192,16, kv_buf,384,32, nullptr,
                                                           att,768,589824, 768,768,16, 0.14433756729740643f);

    // 5. fused bias + point term + mask + weighted softmax (in-place)
    attn_softmax<<<dim3(768,12), 256, 0, stream>>>(att, bbias, qpts, kvppts, hwts, mask, ss);

    // 6. o = a@v (batched per head)
    wmma_gemm<false><<<tiles(768,16,12), blk, 0, stream>>>(att,768,589824, kv_buf+16,384,32, nullptr,
                                                           o_buf,192,16, 768,16,768, 1.f);
    // 7. o_pt = a @ v_pts (batched per head, N=24 edge-clamped)
    wmma_gemm<false><<<tiles(768,24,12), blk, 0, stream>>>(att,768,589824, kvppts+12,432,36, nullptr,
                                                           opt_buf,288,24, 768,24,768, 1.f);
    // 8. o_pair = a . z, batched over residue i (M=12 edge-clamped) -- second z
    //    pass, written straight into feats[:, 576:2112]
    wmma_gemm<false><<<tiles(12,128,768), blk, 0, stream>>>(att,589824,768, z,128,98304, nullptr,
                                                            feats+576,128,2112, 12,128,768, 1.f);

    // 9. assemble feats
    copy_o<<<(768*192+255)/256, 256, 0, stream>>>(o_buf, feats);
    finalize_opt<<<(768*96+255)/256, 256, 0, stream>>>(opt_buf, rot, trans, feats);

    // 10. output projection: (768 x 2112) @ (2112 x 384) + b_out
    wmma_gemm<false><<<tiles(768,384,1), blk, 0, stream>>>(feats,2112,0, w_out,384,0, b_out,
                                                           out,384,0, 768,384,2112, 1.f);
}